// GraphSearchPolicy_30030411333995
// MI455X (gfx1250) — compile-verified
//
#include <hip/hip_runtime.h>
#include <hip/hip_bf16.h>

typedef __attribute__((ext_vector_type(2))) float v2f;
typedef __attribute__((ext_vector_type(8))) float v8f;

#define B_   2048
#define A_   200000
#define K_   256
#define ED_  200
#define AD_  400
#define IN_  600

// ---------------------------------------------------------------------------
// Kernel 1: build X = [ent_table[e] | path_h | rel_table[q]]  -> [B, 600]
// float4 chunks: 150 per row; segment boundaries (200, 400) are chunk-aligned.
// ---------------------------------------------------------------------------
__global__ __launch_bounds__(256) void build_x_kernel(
    const float* __restrict__ path_h, const int* __restrict__ e,
    const int* __restrict__ q, const float* __restrict__ ent,
    const float* __restrict__ rel, float* __restrict__ X)
{
    const int b = blockIdx.x;
    const int c = threadIdx.x;          // chunk index, 0..149 active
    if (c >= 150) return;
    const float4* erow = (const float4*)(ent + (size_t)e[b] * ED_);
    const float4* hrow = (const float4*)(path_h + (size_t)b * ED_);
    const float4* rrow = (const float4*)(rel + (size_t)q[b] * ED_);
    float4 v = (c < 50) ? erow[c]
             : (c < 100) ? hrow[c - 50]
             : rrow[c - 100];
    ((float4*)(X + (size_t)b * IN_))[c] = v;
}

// ---------------------------------------------------------------------------
// Kernel 2/3: Y = act( Ain @ W^T + bias ), V_WMMA_F32_16X16X4_F32.
// Each wave: 32(M) x 16(N) tile, two accumulators sharing one B load.
// Software-pipelined: tiles for step kk+4 are loaded before the WMMAs of kk,
// so multiple global_load_b64 stay in flight across iterations.
// ---------------------------------------------------------------------------
__global__ __launch_bounds__(256) void gemm_wmma_kernel(
    const float* __restrict__ Ain, const float* __restrict__ W,
    const float* __restrict__ bias, float* __restrict__ Y,
    int M, int N, int Kd, int do_relu)
{
    const int lane = threadIdx.x & 31;
    const int wave = threadIdx.x >> 5;
    const int Nt = N >> 4;
    const int tiles = (M >> 5) * Nt;
    const int wg = blockIdx.x * 8 + wave;
    if (wg >= tiles) return;            // whole-wave uniform exit (EXEC all-1)

    const int tm = wg / Nt;
    const int tn = wg - tm * Nt;
    const int m0 = tm << 5;
    const int n0 = tn << 4;

    const int ml = lane & 15;
    const int ks = (lane >> 4) << 1;    // 0 or 2

    const float* a0p = Ain + (size_t)(m0 + ml) * Kd + ks;
    const float* a1p = Ain + (size_t)(m0 + 16 + ml) * Kd + ks;
    const float* bp  = W   + (size_t)(n0 + ml) * Kd + ks;

    v8f c0 = {};
    v8f c1 = {};

    // prologue: load first k-slice
    v2f a0 = *(const v2f*)(a0p);
    v2f a1 = *(const v2f*)(a1p);
    v2f bb = *(const v2f*)(bp);

    int kk = 0;
    for (; kk + 4 < Kd; kk += 4) {
        // prefetch next slice before consuming current one
        v2f na0 = *(const v2f*)(a0p + kk + 4);
        v2f na1 = *(const v2f*)(a1p + kk + 4);
        v2f nbb = *(const v2f*)(bp  + kk + 4);
        c0 = __builtin_amdgcn_wmma_f32_16x16x4_f32(
                false, a0, false, bb, (short)0, c0, false, false);
        c1 = __builtin_amdgcn_wmma_f32_16x16x4_f32(
                false, a1, false, bb, (short)0, c1, false, false);
        a0 = na0; a1 = na1; bb = nbb;
    }
    c0 = __builtin_amdgcn_wmma_f32_16x16x4_f32(
            false, a0, false, bb, (short)0, c0, false, false);
    c1 = __builtin_amdgcn_wmma_f32_16x16x4_f32(
            false, a1, false, bb, (short)0, c1, false, false);

    const int col = n0 + ml;
    const float bv = bias[col];
    const int rbase = m0 + ((lane >> 4) << 3);   // +8 for lanes 16..31
#pragma unroll
    for (int r = 0; r < 8; ++r) {
        float v0 = c0[r] + bv;
        float v1 = c1[r] + bv;
        if (do_relu) { v0 = fmaxf(v0, 0.f); v1 = fmaxf(v1, 0.f); }
        Y[(size_t)(rbase + r) * N + col]      = v0;
        Y[(size_t)(rbase + 16 + r) * N + col] = v1;
    }
}

// ---------------------------------------------------------------------------
// Kernel 4: preds[b,k] = dot( [rel[rs[a]] | ent[es[a]]], X2[b] ).
// One block per b (8 waves), each wave handles 32 k's cooperatively.
// Action row = 100 float4 chunks (rel: 0..49, ent: 50..99; boundary aligned).
// Lane handles chunks lane+32t, t=0..3 -> 4 global_load_b128 per dot,
// each instruction pulling 512B coalesced from the L2-resident tables.
// X2[b] pre-staged in 4 float4 registers per lane; 5-step shuffle reduction.
// ---------------------------------------------------------------------------
__global__ __launch_bounds__(256) void score_kernel(
    const float* __restrict__ X2, const int* __restrict__ keys,
    const int* __restrict__ rs, const int* __restrict__ es,
    const float* __restrict__ rel, const float* __restrict__ ent,
    float* __restrict__ out)
{
    const int b = blockIdx.x;
    const int lane = threadIdx.x & 31;
    const int wave = threadIdx.x >> 5;

    // Pre-stage X2[b] chunks for this lane: chunk = lane + 32*t, t=0..3
    const float4* x2row = (const float4*)(X2 + (size_t)b * AD_);
    float4 x2r[4];
#pragma unroll
    for (int t = 0; t < 4; ++t) {
        const int c = lane + (t << 5);
        if (c < 100) x2r[t] = x2row[c];
        else         x2r[t] = make_float4(0.f, 0.f, 0.f, 0.f);
    }

    const int* krow = keys + (size_t)b * K_;
    for (int k = wave; k < K_; k += 8) {
        const int a  = krow[k];
        const int r  = rs[a];
        const int e2 = es[a];
        const float4* rrow = (const float4*)(rel + (size_t)r  * ED_);
        const float4* erow = (const float4*)(ent + (size_t)e2 * ED_);

        float p = 0.f;
#pragma unroll
        for (int t = 0; t < 4; ++t) {
            const int c = lane + (t << 5);
            if (c < 100) {
                const float4 v = (c < 50) ? rrow[c] : erow[c - 50];
                p = fmaf(v.x, x2r[t].x, p);
                p = fmaf(v.y, x2r[t].y, p);
                p = fmaf(v.z, x2r[t].z, p);
                p = fmaf(v.w, x2r[t].w, p);
            }
        }
        // wave32 butterfly reduction
#pragma unroll
        for (int off = 16; off > 0; off >>= 1)
            p += __shfl_xor(p, off, 32);
        if (lane == 0)
            out[(size_t)b * K_ + k] = p;
    }
}

// ---------------------------------------------------------------------------
extern "C" void kernel_launch(void* const* d_in, const int* in_sizes, int n_in,
                              void* d_out, int out_size, void* d_ws, size_t ws_size,
                              hipStream_t stream) {
    const float* path_h = (const float*)d_in[0];
    const int*   e      = (const int*)d_in[1];
    const int*   q      = (const int*)d_in[2];
    const int*   rs     = (const int*)d_in[3];
    const int*   es     = (const int*)d_in[4];
    const int*   keys   = (const int*)d_in[5];
    const float* ent    = (const float*)d_in[6];
    const float* rel    = (const float*)d_in[7];
    const float* W1     = (const float*)d_in[8];
    const float* b1     = (const float*)d_in[9];
    const float* W2     = (const float*)d_in[10];
    const float* b2     = (const float*)d_in[11];
    float* out = (float*)d_out;

    float* X  = (float*)d_ws;                       // [B, 600]
    float* H1 = X  + (size_t)B_ * IN_;              // [B, 400]
    float* X2 = H1 + (size_t)B_ * AD_;              // [B, 400]

    build_x_kernel<<<B_, 256, 0, stream>>>(path_h, e, q, ent, rel, X);

    // H1 = relu(X @ W1^T + b1): M=2048, N=400, K=600 -> 1600 wave-tiles = 200 blocks
    gemm_wmma_kernel<<<200, 256, 0, stream>>>(X, W1, b1, H1, B_, AD_, IN_, 1);
    // X2 = H1 @ W2^T + b2: M=2048, N=400, K=400
    gemm_wmma_kernel<<<200, 256, 0, stream>>>(H1, W2, b2, X2, B_, AD_, AD_, 0);

    score_kernel<<<B_, 256, 0, stream>>>(X2, keys, rs, es, rel, ent, out);
}